// GATModel_1391569404375
// MI455X (gfx1250) — compile-verified
//
#include <hip/hip_runtime.h>
#include <math.h>

// ---------------------------------------------------------------------------
// GAT 2-layer forward for MI455X (gfx1250, wave32).
// Dense GEMMs use V_WMMA_F32_16X16X4_F32 (full f32 precision matrix core),
// 4 (resp. 3) output tiles per wave sharing one A fragment per K-step so the
// inner loop is branch-free with no EXEC manipulation.
// Edge softmax/aggregation uses float atomics (MAX via monotone uint encode).
// ---------------------------------------------------------------------------

typedef __attribute__((ext_vector_type(2))) float v2f;
typedef __attribute__((ext_vector_type(8))) float v8f;

#define N_NODES 50000
#define F_IN    256
#define HC      512   // H*C
#define H1      8
#define C1      64
#define OUT_C   40
#define NEG_SLOPE 0.2f

// Encoded -inf for the order-preserving float->uint max encoding:
// enc(u) = sign ? ~u : (u | 0x80000000).  -inf = 0xFF800000 -> ~ = 0x007FFFFF
#define ENC_NEG_INF 0x007FFFFFu

__device__ __forceinline__ float lrelu(float x) {
  return x > 0.f ? x : NEG_SLOPE * x;
}

__device__ __forceinline__ void atomicMaxFloatEnc(unsigned* addr, float val) {
  unsigned u = __float_as_uint(val);
  unsigned enc = (u & 0x80000000u) ? ~u : (u | 0x80000000u);
  atomicMax(addr, enc);
}

__device__ __forceinline__ float decMaxFloat(unsigned enc) {
  unsigned u = (enc & 0x80000000u) ? (enc & 0x7FFFFFFFu) : ~enc;
  return __uint_as_float(u);
}

__device__ __forceinline__ v8f wmma_f32(v2f a, v2f b, v8f c) {
  return __builtin_amdgcn_wmma_f32_16x16x4_f32(
      /*neg_a=*/false, a, /*neg_b=*/false, b,
      /*c_mod=*/(short)0, c, /*reuse_a=*/false, /*reuse_b=*/false);
}

// ---------------------------------------------------------------------------
// Fragment layouts per CDNA5 ISA 7.12.2 (f32, 16x16x4):
//   A 16x4 : lanes 0-15 -> M=lane, v0=K0 v1=K1 ; lanes 16-31 -> v0=K2 v1=K3
//   B 4x16 : v0: lanes0-15 K=0, lanes16-31 K=2 ; v1: K=1 / K=3  (N = lane%16)
//   C/D    : VGPR v: lanes0-15 -> M=v, lanes16-31 -> M=v+8 ; N = lane%16
// ---------------------------------------------------------------------------

// GEMM1: Hl[N,512] = X[N,256] @ W1[256,512].
// 8 waves/block; wave w computes rows [row0,row0+16) x cols [w*64, w*64+64)
// as 4 WMMA tiles sharing one A fragment per K-step.  Branch-free inner loop.
__global__ void gat_gemm1(const float* __restrict__ X,
                          const float* __restrict__ W1,
                          float* __restrict__ Hl) {
  const int lane = threadIdx.x & 31;
  const int half = lane >> 4;        // 0 | 1
  const int l16  = lane & 15;
  const int wave = threadIdx.x >> 5; // 0..7
  const int row0 = blockIdx.x * 16;
  const int col0 = wave * 64;

  v8f acc[4] = {};
  const float* arow = X + (size_t)(row0 + l16) * F_IN;

  for (int k = 0; k < F_IN; k += 4) {
    v2f a;
    a.x = arow[k + half * 2 + 0];
    a.y = arow[k + half * 2 + 1];
    const float* b0p = W1 + (size_t)(k + half * 2) * HC + col0 + l16;
    const float* b1p = b0p + HC;  // next K row
#pragma unroll
    for (int t = 0; t < 4; ++t) {
      v2f b;
      b.x = b0p[t * 16];
      b.y = b1p[t * 16];
      acc[t] = wmma_f32(a, b, acc[t]);
    }
  }

#pragma unroll
  for (int t = 0; t < 4; ++t) {
#pragma unroll
    for (int v = 0; v < 8; ++v) {
      Hl[(size_t)(row0 + v + 8 * half) * HC + col0 + t * 16 + l16] = acc[t][v];
    }
  }
}

// GEMM2: G[N,40] = Hr[N,512] @ W2[512,40].  One wave per block computes a
// 16x48 strip (3 tiles); cols >= 40 handled branch-free via clamp+mask.
__global__ void gat_gemm2(const float* __restrict__ Hr,
                          const float* __restrict__ W2,
                          float* __restrict__ G) {
  const int lane = threadIdx.x & 31;
  const int half = lane >> 4;
  const int l16  = lane & 15;
  const int row0 = blockIdx.x * 16;

  float msk[3];
  int   colc[3];
#pragma unroll
  for (int t = 0; t < 3; ++t) {
    int col = t * 16 + l16;
    msk[t]  = (col < OUT_C) ? 1.f : 0.f;
    colc[t] = (col < OUT_C) ? col : (OUT_C - 1);
  }

  v8f acc[3] = {};
  const float* arow = Hr + (size_t)(row0 + l16) * HC;

  for (int k = 0; k < HC; k += 4) {
    v2f a;
    a.x = arow[k + half * 2 + 0];
    a.y = arow[k + half * 2 + 1];
    const float* brow0 = W2 + (size_t)(k + half * 2) * OUT_C;
    const float* brow1 = brow0 + OUT_C;
#pragma unroll
    for (int t = 0; t < 3; ++t) {
      v2f b;
      b.x = brow0[colc[t]] * msk[t];
      b.y = brow1[colc[t]] * msk[t];
      acc[t] = wmma_f32(a, b, acc[t]);
    }
  }

#pragma unroll
  for (int t = 0; t < 3; ++t) {
    if (msk[t] != 0.f) {
#pragma unroll
      for (int v = 0; v < 8; ++v) {
        G[(size_t)(row0 + v + 8 * half) * OUT_C + t * 16 + l16] = acc[t][v];
      }
    }
  }
}

// alpha_{s,d}[n,h] = sum_c Hl[n,h,c] * a_{src,dst}[h,c]
__global__ void gat_alpha1(const float* __restrict__ Hl,
                           const float* __restrict__ a_src,
                           const float* __restrict__ a_dst,
                           float* __restrict__ as, float* __restrict__ ad) {
  int idx = blockIdx.x * blockDim.x + threadIdx.x;
  if (idx >= N_NODES * H1) return;
  int n = idx >> 3, h = idx & 7;
  const float* hv = Hl + (size_t)n * HC + h * C1;
  const float* sv = a_src + h * C1;
  const float* dv = a_dst + h * C1;
  float ss = 0.f, dd = 0.f;
  for (int c = 0; c < C1; ++c) { float v = hv[c]; ss += v * sv[c]; dd += v * dv[c]; }
  as[idx] = ss; ad[idx] = dd;
}

__global__ void fill_u32(unsigned* __restrict__ p, unsigned v, int n) {
  int i = blockIdx.x * blockDim.x + threadIdx.x;
  if (i < n) p[i] = v;
}

// ------------------------- layer 1 edge passes ------------------------------
__global__ void gat_edge_max1(const int* __restrict__ ei, int E,
                              const float* __restrict__ as,
                              const float* __restrict__ ad,
                              unsigned* __restrict__ m1) {
  int idx = blockIdx.x * blockDim.x + threadIdx.x;
  if (idx >= (E + N_NODES) * H1) return;
  int e = idx >> 3, h = idx & 7;
  int s, d;
  if (e < E) { s = ei[e]; d = ei[E + e]; } else { s = d = e - E; }
  float lg = lrelu(as[s * H1 + h] + ad[d * H1 + h]);
  atomicMaxFloatEnc(&m1[d * H1 + h], lg);
}

__global__ void gat_edge_sum1(const int* __restrict__ ei, int E,
                              const float* __restrict__ as,
                              const float* __restrict__ ad,
                              const unsigned* __restrict__ m1,
                              float* __restrict__ sw1) {
  int idx = blockIdx.x * blockDim.x + threadIdx.x;
  if (idx >= (E + N_NODES) * H1) return;
  int e = idx >> 3, h = idx & 7;
  int s, d;
  if (e < E) { s = ei[e]; d = ei[E + e]; } else { s = d = e - E; }
  float lg = lrelu(as[s * H1 + h] + ad[d * H1 + h]);
  float w = expf(lg - decMaxFloat(m1[d * H1 + h]));
  atomicAdd(&sw1[d * H1 + h], w);
}

__global__ void gat_edge_agg1(const int* __restrict__ ei, int E,
                              const float* __restrict__ as,
                              const float* __restrict__ ad,
                              const unsigned* __restrict__ m1,
                              const float* __restrict__ Hl,
                              float* __restrict__ agg) {
  int idx = blockIdx.x * blockDim.x + threadIdx.x;
  if (idx >= (E + N_NODES) * H1) return;
  int e = idx >> 3, h = idx & 7;
  int s, d;
  if (e < E) { s = ei[e]; d = ei[E + e]; } else { s = d = e - E; }
  float lg = lrelu(as[s * H1 + h] + ad[d * H1 + h]);
  float w = expf(lg - decMaxFloat(m1[d * H1 + h]));
  const float* hs = Hl + (size_t)s * HC + h * C1;
  float* ag = agg + (size_t)d * HC + h * C1;
  for (int c = 0; c < C1; ++c) atomicAdd(&ag[c], hs[c] * w);
}

// out = relu(agg/sum + b1), written in place (agg becomes Hr)
__global__ void gat_final1(float* __restrict__ agg,
                           const float* __restrict__ sw1,
                           const float* __restrict__ b1) {
  int idx = blockIdx.x * blockDim.x + threadIdx.x;
  if (idx >= N_NODES * HC) return;
  int n = idx / HC, j = idx % HC;
  float v = agg[idx] / sw1[n * H1 + (j >> 6)] + b1[j];
  agg[idx] = v > 0.f ? v : 0.f;
}

// ------------------------- layer 2 ------------------------------------------
__global__ void gat_alpha2(const float* __restrict__ G,
                           const float* __restrict__ a_s,
                           const float* __restrict__ a_d,
                           float* __restrict__ as, float* __restrict__ ad) {
  int n = blockIdx.x * blockDim.x + threadIdx.x;
  if (n >= N_NODES) return;
  const float* g = G + (size_t)n * OUT_C;
  float ss = 0.f, dd = 0.f;
  for (int j = 0; j < OUT_C; ++j) { float v = g[j]; ss += v * a_s[j]; dd += v * a_d[j]; }
  as[n] = ss; ad[n] = dd;
}

__global__ void gat_edge_max2(const int* __restrict__ ei, int E,
                              const float* __restrict__ as,
                              const float* __restrict__ ad,
                              unsigned* __restrict__ m2) {
  int e = blockIdx.x * blockDim.x + threadIdx.x;
  if (e >= E + N_NODES) return;
  int s, d;
  if (e < E) { s = ei[e]; d = ei[E + e]; } else { s = d = e - E; }
  atomicMaxFloatEnc(&m2[d], lrelu(as[s] + ad[d]));
}

__global__ void gat_edge_sum2(const int* __restrict__ ei, int E,
                              const float* __restrict__ as,
                              const float* __restrict__ ad,
                              const unsigned* __restrict__ m2,
                              float* __restrict__ sw2) {
  int e = blockIdx.x * blockDim.x + threadIdx.x;
  if (e >= E + N_NODES) return;
  int s, d;
  if (e < E) { s = ei[e]; d = ei[E + e]; } else { s = d = e - E; }
  float w = expf(lrelu(as[s] + ad[d]) - decMaxFloat(m2[d]));
  atomicAdd(&sw2[d], w);
}

__global__ void gat_edge_agg2(const int* __restrict__ ei, int E,
                              const float* __restrict__ as,
                              const float* __restrict__ ad,
                              const unsigned* __restrict__ m2,
                              const float* __restrict__ G,
                              float* __restrict__ agg2) {
  int e = blockIdx.x * blockDim.x + threadIdx.x;
  if (e >= E + N_NODES) return;
  int s, d;
  if (e < E) { s = ei[e]; d = ei[E + e]; } else { s = d = e - E; }
  float w = expf(lrelu(as[s] + ad[d]) - decMaxFloat(m2[d]));
  const float* gs = G + (size_t)s * OUT_C;
  float* ag = agg2 + (size_t)d * OUT_C;
  for (int j = 0; j < OUT_C; ++j) atomicAdd(&ag[j], gs[j] * w);
}

// v = agg2/sum + b2 ; out = log_softmax(v)
__global__ void gat_final2(const float* __restrict__ agg2,
                           const float* __restrict__ sw2,
                           const float* __restrict__ b2,
                           float* __restrict__ out) {
  int n = blockIdx.x * blockDim.x + threadIdx.x;
  if (n >= N_NODES) return;
  float inv = 1.f / sw2[n];
  float v[OUT_C];
  float mx = -INFINITY;
  for (int j = 0; j < OUT_C; ++j) {
    v[j] = agg2[(size_t)n * OUT_C + j] * inv + b2[j];
    mx = fmaxf(mx, v[j]);
  }
  float s = 0.f;
  for (int j = 0; j < OUT_C; ++j) s += expf(v[j] - mx);
  float lse = logf(s);
  for (int j = 0; j < OUT_C; ++j) out[(size_t)n * OUT_C + j] = v[j] - mx - lse;
}

// ---------------------------------------------------------------------------
extern "C" void kernel_launch(void* const* d_in, const int* in_sizes, int n_in,
                              void* d_out, int out_size, void* d_ws, size_t ws_size,
                              hipStream_t stream) {
  const float* x      = (const float*)d_in[0];
  const int*   ei     = (const int*)d_in[1];
  const float* W1     = (const float*)d_in[2];
  const float* a_src1 = (const float*)d_in[3];
  const float* a_dst1 = (const float*)d_in[4];
  const float* b1     = (const float*)d_in[5];
  const float* W2     = (const float*)d_in[6];
  const float* a_src2 = (const float*)d_in[7];
  const float* a_dst2 = (const float*)d_in[8];
  const float* b2     = (const float*)d_in[9];
  float* out = (float*)d_out;

  const int E  = in_sizes[1] / 2;
  const int ET = E + N_NODES;

  // ---- workspace layout (floats) ----
  float*    hlin1 = (float*)d_ws;                             // N*512
  float*    agg1  = hlin1 + (size_t)N_NODES * HC;             // N*512 (becomes Hr)
  float*    as1   = agg1 + (size_t)N_NODES * HC;              // N*8
  float*    ad1   = as1 + N_NODES * H1;                       // N*8
  float*    sw1   = ad1 + N_NODES * H1;                       // N*8
  unsigned* m1    = (unsigned*)(sw1 + N_NODES * H1);          // N*8
  float*    g2    = (float*)(m1 + N_NODES * H1);              // N*40
  float*    agg2  = g2 + (size_t)N_NODES * OUT_C;             // N*40
  float*    as2   = agg2 + (size_t)N_NODES * OUT_C;           // N
  float*    ad2   = as2 + N_NODES;                            // N
  float*    sw2   = ad2 + N_NODES;                            // N
  unsigned* m2    = (unsigned*)(sw2 + N_NODES);               // N

  const int TB = 256;

  // ---- layer 1 ----
  gat_gemm1<<<N_NODES / 16, 256, 0, stream>>>(x, W1, hlin1);

  gat_alpha1<<<(N_NODES * H1 + TB - 1) / TB, TB, 0, stream>>>(hlin1, a_src1, a_dst1, as1, ad1);

  hipMemsetAsync(agg1, 0, (size_t)N_NODES * HC * sizeof(float), stream);
  hipMemsetAsync(sw1, 0, (size_t)N_NODES * H1 * sizeof(float), stream);
  fill_u32<<<(N_NODES * H1 + TB - 1) / TB, TB, 0, stream>>>(m1, ENC_NEG_INF, N_NODES * H1);

  {
    int tot = ET * H1;
    int blocks = (tot + TB - 1) / TB;
    gat_edge_max1<<<blocks, TB, 0, stream>>>(ei, E, as1, ad1, m1);
    gat_edge_sum1<<<blocks, TB, 0, stream>>>(ei, E, as1, ad1, m1, sw1);
    gat_edge_agg1<<<blocks, TB, 0, stream>>>(ei, E, as1, ad1, m1, hlin1, agg1);
  }

  gat_final1<<<((size_t)N_NODES * HC + TB - 1) / TB, TB, 0, stream>>>(agg1, sw1, b1);

  // ---- layer 2 ----
  gat_gemm2<<<N_NODES / 16, 32, 0, stream>>>(agg1, W2, g2);

  gat_alpha2<<<(N_NODES + TB - 1) / TB, TB, 0, stream>>>(g2, a_src2, a_dst2, as2, ad2);

  hipMemsetAsync(agg2, 0, (size_t)N_NODES * OUT_C * sizeof(float), stream);
  hipMemsetAsync(sw2, 0, (size_t)N_NODES * sizeof(float), stream);
  fill_u32<<<(N_NODES + TB - 1) / TB, TB, 0, stream>>>(m2, ENC_NEG_INF, N_NODES);

  {
    int blocks = (ET + TB - 1) / TB;
    gat_edge_max2<<<blocks, TB, 0, stream>>>(ei, E, as2, ad2, m2);
    gat_edge_sum2<<<blocks, TB, 0, stream>>>(ei, E, as2, ad2, m2, sw2);
    gat_edge_agg2<<<blocks, TB, 0, stream>>>(ei, E, as2, ad2, m2, g2, agg2);
  }

  gat_final2<<<(N_NODES + TB - 1) / TB, TB, 0, stream>>>(agg2, sw2, b2, out);
}